// ForwardWarpStereo_66846870995014
// MI455X (gfx1250) — compile-verified
//
#include <hip/hip_runtime.h>
#include <hip/hip_bf16.h>
#include <stdint.h>

// Problem dims (from reference setup_inputs): im [16,3,512,960], disp [16,1,512,960]
#define BATCH    16
#define CHANNELS 3
#define HEIGHT   512
#define WIDTH    960
#define EPS_F    1e-6f
// log2(1.414)
#define LOG2_BASE 0.49978680433265427f

// ---------------- CDNA5 async global->LDS DMA (ASYNCcnt path) ----------------
// Stages 16B per lane from global memory into LDS without touching VGPR data.
// Generic LDS addresses on gfx1250 carry the LDS byte offset in addr[31:0]
// (aperture in the high bits), so the low 32 bits are the VDST LDS address.
__device__ __forceinline__ void async_copy16(const float* __restrict__ g, float* l) {
#if defined(__gfx1250__)
  unsigned loff = (unsigned)(uintptr_t)l;
  asm volatile("global_load_async_to_lds_b128 %0, %1, off"
               :: "v"(loff), "v"(g)
               : "memory");
#else
  *(float4*)l = *(const float4*)g;
#endif
}

__device__ __forceinline__ void wait_async_zero() {
#if defined(__gfx1250__)
  asm volatile("s_wait_asynccnt 0x0" ::: "memory");
#endif
}

__device__ __forceinline__ float fast_exp2(float x) {
#if __has_builtin(__builtin_amdgcn_exp2f)
  return __builtin_amdgcn_exp2f(x);   // v_exp_f32; args here are in [0, ~20]
#else
  return exp2f(x);
#endif
}

// ---------------- Kernel 1: exact global min of disp ----------------
// disp >= 0 for this workload, so float bit patterns order like unsigned ints.
// Launched with grid*block such that n/4 == threads*8 for the fast path:
// 8 independent b128 loads in flight per wave instead of load->wait->load.
__global__ __launch_bounds__(256) void disp_min_kernel(
    const float* __restrict__ disp, unsigned* __restrict__ minbits, int n) {
  __shared__ float red[256];
  const int tid    = threadIdx.x;
  const int stride = gridDim.x * blockDim.x;
  const int gid    = blockIdx.x * blockDim.x + tid;
  float m = 3.4e38f;

  const int n4 = n >> 2;
  const float4* __restrict__ d4 = (const float4*)disp;

  if (n4 == stride * 8) {
    // Fast path: exactly 8 strided float4 loads per thread, all independent.
    float4 v[8];
    #pragma unroll
    for (int k = 0; k < 8; ++k) v[k] = d4[gid + k * stride];
    #pragma unroll
    for (int k = 0; k < 8; ++k)
      m = fminf(m, fminf(fminf(v[k].x, v[k].y), fminf(v[k].z, v[k].w)));
  } else {
    for (int i = gid; i < n4; i += stride) {
      float4 v = d4[i];
      m = fminf(m, fminf(fminf(v.x, v.y), fminf(v.z, v.w)));
    }
    for (int i = (n4 << 2) + gid; i < n; i += stride)
      m = fminf(m, disp[i]);
  }

  red[tid] = m;
  __syncthreads();
  #pragma unroll
  for (int s = 128; s > 0; s >>= 1) {
    if (tid < s) red[tid] = fminf(red[tid], red[tid + s]);
    __syncthreads();
  }
  if (tid == 0) atomicMin(minbits, __float_as_uint(red[0]));
}

// ---------------- Kernel 2: per-row forward splat + normalize ----------------
// flow_y == 0  =>  each source pixel scatters only within its own row, to
// x0 = floor(x - d) and x0+1. One workgroup per (b, y) row; accumulate the
// 5 channels in LDS with ds_add_f32 atomics, then normalize and store b128.
__global__ __launch_bounds__(256) void splat_row_kernel(
    const float* __restrict__ im, const float* __restrict__ disp,
    const unsigned* __restrict__ minbits, float* __restrict__ out) {
  __shared__ __align__(16) float s_disp[WIDTH];
  __shared__ __align__(16) float s_im[CHANNELS][WIDTH];
  __shared__ __align__(16) float s_acc[5][WIDTH];

  const int tid = threadIdx.x;
  const int row = blockIdx.x;            // 0 .. B*H-1
  const int b   = row / HEIGHT;
  const int y   = row - b * HEIGHT;

  const size_t row_off = ((size_t)b * HEIGHT + y) * WIDTH;
  const float* __restrict__ gdisp = disp + row_off;
  const float* __restrict__ gim0  = im + (((size_t)b * CHANNELS + 0) * HEIGHT + y) * WIDTH;
  const float* __restrict__ gim1  = gim0 + (size_t)HEIGHT * WIDTH;
  const float* __restrict__ gim2  = gim1 + (size_t)HEIGHT * WIDTH;

  // Issue async DMA of the 4 row slices into LDS (240 x 16B chunks each).
  if (tid < WIDTH / 4) {
    const int e = tid * 4;
    async_copy16(gdisp + e, &s_disp[e]);
    async_copy16(gim0  + e, &s_im[0][e]);
    async_copy16(gim1  + e, &s_im[1][e]);
    async_copy16(gim2  + e, &s_im[2][e]);
  }

  // Zero the accumulators (b128 LDS stores) while the DMA is in flight.
  {
    float4* __restrict__ acc4 = (float4*)&s_acc[0][0];
    const float4 z = make_float4(0.0f, 0.0f, 0.0f, 0.0f);
    #pragma unroll
    for (int i = 0; i < 5; ++i) {
      const int c = tid + i * 256;                 // 5*WIDTH/4 = 1200 chunks
      if (c < 5 * WIDTH / 4) acc4[c] = z;
    }
  }

  const float dmin = __uint_as_float(*minbits);

  wait_async_zero();
  __syncthreads();

  // Scatter phase: 10 ds_add_f32 per source pixel, all WGP-local.
  for (int x = tid; x < WIDTH; x += 256) {
    const float d   = s_disp[x];
    const float wm  = fast_exp2((d - dmin) * LOG2_BASE);   // 1.414^(d - dmin)
    const float tx  = (float)x - d;
    const float x0f = floorf(tx);
    const float wx1 = tx - x0f;
    const float wx0 = 1.0f - wx1;
    const int   x0  = (int)x0f;
    const int   x1  = x0 + 1;

    const float v0 = s_im[0][x] * wm;
    const float v1 = s_im[1][x] * wm;
    const float v2 = s_im[2][x] * wm;

    if (x0 >= 0 && x0 < WIDTH) {
      atomicAdd(&s_acc[0][x0], v0 * wx0);
      atomicAdd(&s_acc[1][x0], v1 * wx0);
      atomicAdd(&s_acc[2][x0], v2 * wx0);
      atomicAdd(&s_acc[3][x0], wm * wx0);
      atomicAdd(&s_acc[4][x0], wx0);
    }
    if (x1 >= 0 && x1 < WIDTH) {
      atomicAdd(&s_acc[0][x1], v0 * wx1);
      atomicAdd(&s_acc[1][x1], v1 * wx1);
      atomicAdd(&s_acc[2][x1], v2 * wx1);
      atomicAdd(&s_acc[3][x1], wm * wx1);
      atomicAdd(&s_acc[4][x1], wx1);
    }
  }
  __syncthreads();

  // Normalize + write out as b128: res [B,3,H,W] then occ [B,1,H,W] (concat).
  // 240 float4 chunks per row -> single pass of the 256-thread block.
  float* __restrict__ out_res = out;
  float* __restrict__ out_occ = out + (size_t)BATCH * CHANNELS * HEIGHT * WIDTH;
  const size_t res0  = (((size_t)b * CHANNELS + 0) * HEIGHT + y) * WIDTH;
  const size_t plane = (size_t)HEIGHT * WIDTH;

  if (tid < WIDTH / 4) {
    const int o = tid * 4;
    const float4 a0 = *(const float4*)&s_acc[0][o];
    const float4 a1 = *(const float4*)&s_acc[1][o];
    const float4 a2 = *(const float4*)&s_acc[2][o];
    const float4 am = *(const float4*)&s_acc[3][o];
    const float4 aw = *(const float4*)&s_acc[4][o];

    float4 i4;
    i4.x = 1.0f / fmaxf(am.x, EPS_F);
    i4.y = 1.0f / fmaxf(am.y, EPS_F);
    i4.z = 1.0f / fmaxf(am.z, EPS_F);
    i4.w = 1.0f / fmaxf(am.w, EPS_F);

    float4 r;
    r.x = a0.x * i4.x; r.y = a0.y * i4.y; r.z = a0.z * i4.z; r.w = a0.w * i4.w;
    *(float4*)&out_res[res0 + o] = r;
    r.x = a1.x * i4.x; r.y = a1.y * i4.y; r.z = a1.z * i4.z; r.w = a1.w * i4.w;
    *(float4*)&out_res[res0 + plane + o] = r;
    r.x = a2.x * i4.x; r.y = a2.y * i4.y; r.z = a2.z * i4.z; r.w = a2.w * i4.w;
    *(float4*)&out_res[res0 + 2 * plane + o] = r;

    float4 oc;
    oc.x = 1.0f - fminf(fmaxf(aw.x, 0.0f), 1.0f);
    oc.y = 1.0f - fminf(fmaxf(aw.y, 0.0f), 1.0f);
    oc.z = 1.0f - fminf(fmaxf(aw.z, 0.0f), 1.0f);
    oc.w = 1.0f - fminf(fmaxf(aw.w, 0.0f), 1.0f);
    *(float4*)&out_occ[row_off + o] = oc;
  }
}

extern "C" void kernel_launch(void* const* d_in, const int* in_sizes, int n_in,
                              void* d_out, int out_size, void* d_ws, size_t ws_size,
                              hipStream_t stream) {
  const float* im   = (const float*)d_in[0];
  const float* disp = (const float*)d_in[1];
  float* out        = (float*)d_out;
  unsigned* minbits = (unsigned*)d_ws;
  const int n_disp  = in_sizes[1];   // B*H*W = 7,864,320

  // 0x7f7f7f7f as float bits ~= 3.39e38 > any disp; async memset is graph-capturable.
  hipMemsetAsync(d_ws, 0x7f, sizeof(unsigned), stream);

  // 960 blocks * 256 threads * 8 float4 each == n_disp/4 -> fast path in kernel.
  disp_min_kernel<<<960, 256, 0, stream>>>(disp, minbits, n_disp);
  splat_row_kernel<<<BATCH * HEIGHT, 256, 0, stream>>>(im, disp, minbits, out);
}